// DisLayer_12756052869807
// MI455X (gfx1250) — compile-verified
//
#include <hip/hip_runtime.h>
#include <stdint.h>

// Problem dims (fixed by the reference)
#define Bz 16
#define Cz 256
#define Wz 112
#define Hz 112

// LDS tile geometry: one (b,c) half-image strip of 56 output rows per block.
#define XROWS 64    // 56 + 4 halo top + 4 halo bottom
#define XSTR  124   // padded LDS row stride in floats (keeps interior 16B-aligned)
#define VROWS 60    // 56 + 2 + 2 (conv1 output incl. halo for conv2)
#define VSTR  120   // 116 cols + pad

typedef uint32_t u32x4 __attribute__((ext_vector_type(4)));
typedef uint32_t u32x8 __attribute__((ext_vector_type(8)));
typedef float    f32x2 __attribute__((ext_vector_type(2)));
typedef float    f32x4 __attribute__((ext_vector_type(4)));

// ---------------------------------------------------------------------------
// Kernel 1: mean Gaussian-pdf map m[112*112] (tiny; stays hot in 192MB L2)
// ---------------------------------------------------------------------------
__global__ void __launch_bounds__(256) pdf_map_kernel(
    const float* __restrict__ loc, const float* __restrict__ scal,
    const float* __restrict__ ps, float* __restrict__ m) {
  int idx = blockIdx.x * 256 + threadIdx.x;
  if (idx >= Wz * Hz) return;
  float p  = ps[0];
  float x0 = (float)(idx / Hz) * p;
  float x1 = (float)(idx % Hz) * p;
  float acc = 0.0f;
#pragma unroll
  for (int l = 0; l < 8; ++l) {
    float s0 = scal[2 * l + 0], s1 = scal[2 * l + 1];
    float d0 = (x0 - loc[2 * l + 0]) / s0;
    float d1 = (x1 - loc[2 * l + 1]) / s1;
    float lp = -0.5f * (d0 * d0 + d1 * d1) - __logf(s0) - __logf(s1)
               - 1.8378770664093455f; // log(2*pi)
    acc += __expf(lp);
  }
  m[idx] = acc * 0.125f; // mean over L=8
}

// ---------------------------------------------------------------------------
// 4 adjacent outputs of a depthwise 5x5 from an LDS tile.
// Even taps (dx=0,2,4) are 8B-aligned float2 pairs -> guaranteed v_pk_fma_f32
// (12 of 20 FMAs per row packed); odd taps are scalar FMAs on register halves
// (VOPD dual-issue fodder). Loads: 4x ds_load_b64 per row instead of 8x b32.
// ---------------------------------------------------------------------------
__device__ __forceinline__ void dw5x4(const float* __restrict__ src, int stride,
                                      const float* __restrict__ w, float bias,
                                      float a[4]) {
  f32x2 acc01 = {bias, bias};
  f32x2 acc23 = {bias, bias};
#pragma unroll
  for (int dy = 0; dy < 5; ++dy) {
    const float* r = src + dy * stride;            // 8B-aligned (stride even, col%4==0)
    f32x2 p0 = *(const f32x2*)(r + 0);             // (s0,s1)
    f32x2 p1 = *(const f32x2*)(r + 2);             // (s2,s3)
    f32x2 p2 = *(const f32x2*)(r + 4);             // (s4,s5)
    f32x2 p3 = *(const f32x2*)(r + 6);             // (s6,s7)
    const float w0 = w[5*dy+0], w1 = w[5*dy+1], w2 = w[5*dy+2];
    const float w3 = w[5*dy+3], w4 = w[5*dy+4];
    // even taps: packed
    acc01 += w0 * p0;   acc23 += w0 * p1;
    acc01 += w2 * p1;   acc23 += w2 * p2;
    acc01 += w4 * p2;   acc23 += w4 * p3;
    // odd taps: scalar on halves
    acc01.x += w1 * p0.y;   acc01.y += w1 * p1.x;
    acc23.x += w1 * p1.y;   acc23.y += w1 * p2.x;
    acc01.x += w3 * p1.y;   acc01.y += w3 * p2.x;
    acc23.x += w3 * p2.y;   acc23.y += w3 * p3.x;
  }
  a[0] = acc01.x; a[1] = acc01.y; a[2] = acc23.x; a[3] = acc23.y;
}

// ---------------------------------------------------------------------------
// Kernel 2: fused  out = x + conv2(relu(conv1(x))) * m
// One block per (b, c, strip): blockIdx.x = bc*2 + strip, strip of 56 rows.
// Input strip is DMA'd into LDS by the Tensor Data Mover (iterate mode),
// overlapped with the halo zeroing done by the ALUs.
// ---------------------------------------------------------------------------
__global__ void __launch_bounds__(256) fused_dw_kernel(
    const float* __restrict__ x,  const float* __restrict__ w1g,
    const float* __restrict__ b1g, const float* __restrict__ w2g,
    const float* __restrict__ b2g, const float* __restrict__ m,
    float* __restrict__ out) {
  __shared__ float xt[XROWS * XSTR]; // 31,744 B
  __shared__ float vt[VROWS * VSTR]; // 28,800 B

  const int tid   = threadIdx.x;
  const int blk   = blockIdx.x;
  const int strip = blk & 1;
  const int bc    = blk >> 1;
  const int c     = bc & (Cz - 1);
  const int i0    = strip * 56;
  const size_t plane = (size_t)bc * (Wz * Hz);

  // ---- Phase A1: kick off the TDM DMA of the 60x112 interior ------------
  if (tid == 0) {
    // 60 valid input rows: gStart..gStart+59, landing at LDS tile row ldsRow.
    const int gStart = (i0 == 0) ? 0 : (i0 - 4);
    const int ldsRow = (i0 == 0) ? 4 : 0;
    const uint64_t gaddr = (uint64_t)(uintptr_t)(x + plane + (size_t)gStart * Hz);
    // low 32 bits of a flat LDS pointer == wave-relative LDS byte offset
    const uint32_t ldsa =
        (uint32_t)(uintptr_t)xt + (uint32_t)((ldsRow * XSTR + 4) * 4);

    // Tensor DMA descriptor (CDNA5 ISA),
    // iterate mode: 60 iterations of a 112-float row, LDS stride 124 floats.
    u32x4 g0;
    g0[0] = 1u;                                              // count=1, user mode
    g0[1] = ldsa;                                            // lds_addr (bytes)
    g0[2] = (uint32_t)gaddr;                                 // global_addr[31:0]
    g0[3] = ((uint32_t)(gaddr >> 32) & 0x01FFFFFFu) | (2u << 30); // [56:32], type=2
    u32x8 g1;
    g1[0] = (2u << 16) | (1u << 19);   // data_size=4B, iterate_enable=1
    g1[1] = ((uint32_t)Hz) << 16;      // tensor_dim0 = 112 (bits 79:48, lo16)
    g1[2] = (1u << 16);                // tensor_dim0 hi=0 | tensor_dim1 lo=1
    g1[3] = ((uint32_t)Hz) << 16;      // tensor_dim1 hi=0 | tile_dim0 = 112
    g1[4] = 1u;                        // tile_dim1=1, tile_dim2=0
    g1[5] = (uint32_t)Hz;              // tensor_dim0_stride = 112 (lo32)
    g1[6] = ((uint32_t)Hz) << 16;      // stride0 hi=0 | tensor_dim1_stride lo16
    g1[7] = 0u;                        // tensor_dim1_stride hi
    u32x4 g2;
    g2[0] = 1u;                        // tensor_dim2 (unused)
    g2[1] = (uint32_t)XSTR;            // lds_addr_increment   = 124 elements
    g2[2] = (uint32_t)Hz;              // global_addr_increment= 112 elements
    g2[3] = 59u << 16;                 // iterate_count = 59 -> 60 rows
    u32x4 g3 = {0u, 0u, 0u, 0u};

    asm volatile("tensor_load_to_lds %0, %1, %2, %3"
                 :: "s"(g0), "s"(g1), "s"(g2), "s"(g3)
                 : "memory");
  }

  // Block-uniform weights -> scalar registers (overlaps with DMA)
  float w1r[25], w2r[25];
#pragma unroll
  for (int k = 0; k < 25; ++k) w1r[k] = w1g[c * 25 + k];
#pragma unroll
  for (int k = 0; k < 25; ++k) w2r[k] = w2g[c * 25 + k];
  const float b1v = b1g[c];
  const float b2v = b2g[c];

  // ---- Phase A2: zero-pad halo (disjoint from the DMA'd cells) ----------
  for (int t = tid; t < 960; t += 256) {
    int r, cc;
    if (t < 480) {                       // 4 fully-out-of-image rows
      r  = (i0 == 0 ? 0 : 60) + t / 120;
      cc = t % 120;
    } else {                             // left/right 4-col halos of the 60 loaded rows
      int u = t - 480;
      r  = (i0 == 0 ? 4 : 0) + u / 8;
      int q = u & 7;
      cc = (q < 4) ? q : 112 + q;
    }
    xt[r * XSTR + cc] = 0.0f;
  }

  if (tid == 0) {
    asm volatile("s_wait_tensorcnt 0x0" ::: "memory");
  }
  __syncthreads();

  // ---- Phase B: conv1 + ReLU into vt (zeroing v-halo so conv2's zero
  //      padding matches the reference exactly) ----------------------------
  for (int t = tid; t < VROWS * 29; t += 256) {
    const int ii = t / 29;
    const int jj = (t % 29) * 4;       // vt col, v-coord gj = jj-2..jj+1
    float a[4];
    dw5x4(&xt[ii * XSTR + jj], XSTR, w1r, b1v, a);
    const int gi1 = i0 + ii - 2;
    const bool rowOK = (gi1 >= 0) & (gi1 <= Wz - 1);
#pragma unroll
    for (int q = 0; q < 4; ++q) {
      const int gj = jj - 2 + q;
      const bool ok = rowOK & (gj >= 0) & (gj <= Hz - 1);
      vt[ii * VSTR + jj + q] = ok ? fmaxf(a[q], 0.0f) : 0.0f;
    }
  }
  __syncthreads();

  // ---- Phase C: conv2 + finalize  out = x + v2 * m  (16B vector ops) -----
  for (int t = tid; t < 56 * 28; t += 256) {
    const int ii = t / 28;
    const int jj = (t % 28) * 4;
    const int gi = i0 + ii;
    float a[4];
    dw5x4(&vt[ii * VSTR + jj], VSTR, w2r, b2v, a);
    const f32x4 mm = *reinterpret_cast<const f32x4*>(m + gi * Hz + jj);
    const f32x4 xv = *reinterpret_cast<const f32x4*>(&xt[(ii + 4) * XSTR + (jj + 4)]);
    f32x4 av = {a[0], a[1], a[2], a[3]};
    *reinterpret_cast<f32x4*>(out + plane + (size_t)gi * Hz + jj) = xv + av * mm;
  }
}

// ---------------------------------------------------------------------------
extern "C" void kernel_launch(void* const* d_in, const int* in_sizes, int n_in,
                              void* d_out, int out_size, void* d_ws, size_t ws_size,
                              hipStream_t stream) {
  (void)in_sizes; (void)n_in; (void)out_size; (void)ws_size;
  const float* x     = (const float*)d_in[0];
  const float* w1    = (const float*)d_in[1];
  const float* b1    = (const float*)d_in[2];
  const float* w2    = (const float*)d_in[3];
  const float* b2    = (const float*)d_in[4];
  const float* nloc  = (const float*)d_in[5];
  const float* nscal = (const float*)d_in[6];
  const float* pscal = (const float*)d_in[7];
  float* out = (float*)d_out;
  float* m   = (float*)d_ws;           // 112*112 floats (~50 KB scratch)

  pdf_map_kernel<<<(Wz * Hz + 255) / 256, 256, 0, stream>>>(nloc, nscal, pscal, m);
  fused_dw_kernel<<<Bz * Cz * 2, 256, 0, stream>>>(x, w1, b1, w2, b2, m, out);
}